// Decoder_67851893342414
// MI455X (gfx1250) — compile-verified
//
#include <hip/hip_runtime.h>

typedef float v2f __attribute__((ext_vector_type(2)));
typedef float v8f __attribute__((ext_vector_type(8)));

constexpr int NUM_TOPICS = 50;
constexpr int NUM_TIMES  = 50;
constexpr int RSZ        = 256;
constexpr int VOCAB      = 50000;
constexpr int M_ROWS     = NUM_TOPICS * NUM_TIMES;   // 2500 (row m = t*50 + k)
constexpr int M_TILES    = 157;                      // ceil(2500/16)
constexpr int M_PAD      = M_TILES * 16;             // 2512
constexpr int N_TILES    = VOCAB / 16;               // 3125 (exact)
constexpr int N_FULL_GROUPS = N_TILES / 4;           // 781 full groups of 4 tiles
constexpr int N_TAIL_BASE   = N_FULL_GROUPS * 4;     // 3124 (one remainder tile)
constexpr float DELTA    = 0.005f;
constexpr float LOG_DELTA = -5.2983173665480363f;    // log(0.005)

__global__ void detm_init_kernel(float* kl_out) {
    if (threadIdx.x == 0 && blockIdx.x == 0) *kl_out = 0.0f;
}

// One block per (possibly padded) row m, 256 threads over R.
__global__ __launch_bounds__(256)
void detm_alphas_kl_kernel(const float* __restrict__ mu,
                           const float* __restrict__ ls,
                           const float* __restrict__ eps,
                           float* __restrict__ alphas,
                           float* __restrict__ kl_out) {
    const int m = blockIdx.x;      // 0..M_PAD-1
    const int r = threadIdx.x;     // 0..255
    float a  = 0.0f;
    float kl = 0.0f;
    if (m < M_ROWS) {
        const int t = m / NUM_TOPICS;
        const int k = m % NUM_TOPICS;
        const size_t iq = ((size_t)k * NUM_TIMES + t) * RSZ + r;   // [K,T,R]
        const size_t ie = ((size_t)t * NUM_TOPICS + k) * RSZ + r;  // [T,K,R]
        const float muv = mu[iq];
        const float lsv = ls[iq];
        a = muv + eps[ie] * expf(0.5f * lsv);
        const float sq = expf(lsv);
        if (t == 0) {
            kl = (sq + muv * muv) / (1.0f + 1e-6f) - 1.0f - lsv;
        } else {
            const size_t iqp = ((size_t)k * NUM_TIMES + (t - 1)) * RSZ + r;
            const size_t iep = ((size_t)(t - 1) * NUM_TOPICS + k) * RSZ + r;
            const float pmu = mu[iqp] + eps[iep] * expf(0.5f * ls[iqp]);
            const float d = muv - pmu;
            kl = (sq + d * d) / (DELTA + 1e-6f) - 1.0f + LOG_DELTA - lsv;
        }
    }
    alphas[(size_t)m * RSZ + r] = a;

    __shared__ float red[256];
    red[threadIdx.x] = 0.5f * kl;
    __syncthreads();
    for (int s = 128; s > 0; s >>= 1) {
        if (threadIdx.x < s) red[threadIdx.x] += red[threadIdx.x + s];
        __syncthreads();
    }
    if (threadIdx.x == 0) atomicAdd(kl_out, red[0]);
}

// One block (8 wave32) per 16-row M tile; fused f32-WMMA GEMM over the whole
// vocab with online softmax, raw logits staged in d_out, rescaled in place.
__global__ __launch_bounds__(256)
void detm_gemm_softmax_kernel(const float* __restrict__ alphas,
                              const float* __restrict__ W,
                              float* __restrict__ out) {
    const int m_base = blockIdx.x * 16;
    const int tid  = threadIdx.x;
    const int lane = tid & 31;
    const int wave = tid >> 5;
    const int lr = lane & 15;   // A row / B,D column within tile
    const int lh = lane >> 4;   // half-wave selects K pair / D row group
    const bool full_rows = (m_base + 16 <= M_ROWS);  // false only for last block

    __shared__ float As[16 * RSZ];        // 16 KB staged A tile
    __shared__ float redm[16 * 128];      // per-row partial maxes (8 waves x 16 col-lanes)
    __shared__ float reds[16 * 128];      // per-row partial sums
    __shared__ float rowM[16];
    __shared__ float rowS[16];

    // Stage A rows (zero-padded by producer kernel) into LDS.
    for (int i = tid; i < 16 * RSZ / 4; i += 256) {
        ((float4*)As)[i] = ((const float4*)(alphas + (size_t)m_base * RSZ))[i];
    }
    __syncthreads();

    float mrun[8], srun[8];
#pragma unroll
    for (int v = 0; v < 8; ++v) { mrun[v] = -__builtin_inff(); srun[v] = 0.0f; }

    const float* Arow = As + lr * RSZ;

    // ---- Main sweep: 781 full groups of 4 vocab tiles, branch-free hot loop ----
    for (int g = wave; g < N_FULL_GROUPS; g += 8) {
        const int vbase = g * 64;
        v8f acc0, acc1, acc2, acc3;
#pragma unroll
        for (int v = 0; v < 8; ++v) { acc0[v] = 0.f; acc1[v] = 0.f; acc2[v] = 0.f; acc3[v] = 0.f; }

        const float* Wp0 = W + (size_t)(vbase +  0 + lr) * RSZ;
        const float* Wp1 = W + (size_t)(vbase + 16 + lr) * RSZ;
        const float* Wp2 = W + (size_t)(vbase + 32 + lr) * RSZ;
        const float* Wp3 = W + (size_t)(vbase + 48 + lr) * RSZ;

#pragma unroll 4
        for (int k0 = 0; k0 < RSZ; k0 += 4) {
            const int kk = k0 + 2 * lh;
            // f32 A 16x4 frag: lane holds row lr, K = {kk, kk+1} (LDS)
            v2f a  = *(const v2f*)(Arow + kk);
            // f32 B 4x16 frags: lane holds column lr (= W row), K = {kk, kk+1}
            v2f b0 = *(const v2f*)(Wp0 + kk);
            v2f b1 = *(const v2f*)(Wp1 + kk);
            v2f b2 = *(const v2f*)(Wp2 + kk);
            v2f b3 = *(const v2f*)(Wp3 + kk);
            acc0 = __builtin_amdgcn_wmma_f32_16x16x4_f32(false, a, false, b0, (short)0, acc0, false, false);
            acc1 = __builtin_amdgcn_wmma_f32_16x16x4_f32(false, a, false, b1, (short)0, acc1, false, false);
            acc2 = __builtin_amdgcn_wmma_f32_16x16x4_f32(false, a, false, b2, (short)0, acc2, false, false);
            acc3 = __builtin_amdgcn_wmma_f32_16x16x4_f32(false, a, false, b3, (short)0, acc3, false, false);
        }

        // Online softmax: one rescale per row-stream covering all 4 tile values.
#pragma unroll
        for (int v = 0; v < 8; ++v) {
            const float x0 = acc0[v], x1 = acc1[v], x2 = acc2[v], x3 = acc3[v];
            const float mx = fmaxf(fmaxf(x0, x1), fmaxf(x2, x3));
            const float nm = fmaxf(mrun[v], mx);
            srun[v] = srun[v] * __expf(mrun[v] - nm) +
                      __expf(x0 - nm) + __expf(x1 - nm) +
                      __expf(x2 - nm) + __expf(x3 - nm);
            mrun[v] = nm;
        }
        // Raw logit stores: block-uniform guard; 4 stores per row share a base.
        if (full_rows) {
#pragma unroll
            for (int v = 0; v < 8; ++v) {
                const int row = m_base + v + 8 * lh;
                float* rp = out + (size_t)row * VOCAB + vbase + lr;
                rp[0]  = acc0[v];
                rp[16] = acc1[v];
                rp[32] = acc2[v];
                rp[48] = acc3[v];
            }
        } else {
#pragma unroll
            for (int v = 0; v < 8; ++v) {
                const int row = m_base + v + 8 * lh;
                if (row < M_ROWS) {
                    float* rp = out + (size_t)row * VOCAB + vbase + lr;
                    rp[0]  = acc0[v];
                    rp[16] = acc1[v];
                    rp[32] = acc2[v];
                    rp[48] = acc3[v];
                }
            }
        }
    }

    // ---- Tail: remaining N_TILES % 4 tiles (one), strided over waves ----
    for (int vt = N_TAIL_BASE + wave; vt < N_TILES; vt += 8) {
        const int vb = vt * 16;
        v8f acc;
#pragma unroll
        for (int v = 0; v < 8; ++v) acc[v] = 0.f;
        const float* Wp = W + (size_t)(vb + lr) * RSZ;
#pragma unroll 4
        for (int k0 = 0; k0 < RSZ; k0 += 4) {
            const int kk = k0 + 2 * lh;
            v2f a = *(const v2f*)(Arow + kk);
            v2f b = *(const v2f*)(Wp + kk);
            acc = __builtin_amdgcn_wmma_f32_16x16x4_f32(false, a, false, b, (short)0, acc, false, false);
        }
#pragma unroll
        for (int v = 0; v < 8; ++v) {
            const float x  = acc[v];
            const float nm = fmaxf(mrun[v], x);
            srun[v] = srun[v] * __expf(mrun[v] - nm) + __expf(x - nm);
            mrun[v] = nm;
            const int row = m_base + v + 8 * lh;
            if (row < M_ROWS) out[(size_t)row * VOCAB + vb + lr] = x;
        }
    }

    // Combine per-row (max, sum) partials across 8 waves x 16 column lanes.
#pragma unroll
    for (int v = 0; v < 8; ++v) {
        const int row = v + 8 * lh;
        redm[row * 128 + wave * 16 + lr] = mrun[v];
        reds[row * 128 + wave * 16 + lr] = srun[v];
    }
    __syncthreads();
    if (tid < 16) {
        float mm = -__builtin_inff(), ss = 0.0f;
        for (int i = 0; i < 128; ++i) {
            const float mi = redm[tid * 128 + i];
            const float si = reds[tid * 128 + i];
            const float nm = fmaxf(mm, mi);
            ss = ss * __expf(mm - nm) + si * __expf(mi - nm);
            mm = nm;
        }
        rowM[tid] = mm;
        rowS[tid] = ss;
    }
    __threadfence();   // make this block's logit stores visible to all its waves
    __syncthreads();

    // In-place rescale: beta = exp(logit - max) / sum, float4-vectorized.
    const int nvalid = min(16, M_ROWS - m_base);
    for (int r = 0; r < nvalid; ++r) {
        const float mr  = rowM[r];
        const float inv = 1.0f / rowS[r];
        float4* rowp = (float4*)(out + (size_t)(m_base + r) * VOCAB);
        for (int c = tid; c < VOCAB / 4; c += 256) {
            float4 x = rowp[c];
            x.x = __expf(x.x - mr) * inv;
            x.y = __expf(x.y - mr) * inv;
            x.z = __expf(x.z - mr) * inv;
            x.w = __expf(x.w - mr) * inv;
            rowp[c] = x;
        }
    }
}

extern "C" void kernel_launch(void* const* d_in, const int* in_sizes, int n_in,
                              void* d_out, int out_size, void* d_ws, size_t ws_size,
                              hipStream_t stream) {
    const float* mu  = (const float*)d_in[0];
    const float* ls  = (const float*)d_in[1];
    const float* eps = (const float*)d_in[2];
    const float* W   = (const float*)d_in[3];
    float* out    = (float*)d_out;                       // [2500*50000] beta + [1] kl
    float* alphas = (float*)d_ws;                        // M_PAD * 256 f32 (~2.6 MB)
    float* kl     = out + (size_t)M_ROWS * VOCAB;

    detm_init_kernel<<<1, 1, 0, stream>>>(kl);
    detm_alphas_kl_kernel<<<M_PAD, 256, 0, stream>>>(mu, ls, eps, alphas, kl);
    detm_gemm_softmax_kernel<<<M_TILES, 256, 0, stream>>>(alphas, W, out);
}